// GATEncoder_55946243998123
// MI455X (gfx1250) — compile-verified
//
#include <hip/hip_runtime.h>
#include <hip/hip_bf16.h>

typedef __attribute__((ext_vector_type(2))) float v2f;
typedef __attribute__((ext_vector_type(8))) float v8f;

#define LRELU_SLOPE 0.2f

// ---------- monotone float <-> uint key for atomic max ----------
__device__ __forceinline__ unsigned fkey(float f) {
  unsigned b = __float_as_uint(f);
  return (b & 0x80000000u) ? ~b : (b | 0x80000000u);
}
__device__ __forceinline__ float funkey(unsigned u) {
  unsigned b = (u & 0x80000000u) ? (u & 0x7FFFFFFFu) : ~u;
  return __uint_as_float(b);
}

// ---------- fp32 WMMA GEMM: C[M,N] = A[M,K] * B[K,N] ----------
// wave computes a 16x64 tile: 4 accumulators, A fragment reused 4x.
// A frag (16x4 f32): lanes 0-15 rows, v0/v1 = K+0/K+1; lanes 16-31 = K+2/K+3.
// B frag (4x16 f32): lanes 0-15 cols @ K+0/K+1; lanes 16-31 cols @ K+2/K+3.
// C/D (16x16 f32): VGPR v -> row v (lanes 0-15) / row v+8 (lanes 16-31).
__global__ void gat_wmma_gemm_f32(const float* __restrict__ A,
                                  const float* __restrict__ B,
                                  float* __restrict__ C,
                                  int M, int K, int N) {
  int wave = (blockIdx.x * blockDim.x + threadIdx.x) >> 5;
  int lane = threadIdx.x & 31;
  int n64 = N >> 6;
  int mt = wave / n64;
  int nb = wave % n64;
  int mtiles = (M + 15) >> 4;
  if (mt >= mtiles) return;

  int row  = mt * 16 + (lane & 15);
  int rowc = row < M ? row : M - 1;
  int khalf = (lane >> 4) << 1;   // 0 or 2
  int colb  = nb * 64 + (lane & 15);

  v8f acc[4];
#pragma unroll
  for (int j = 0; j < 4; ++j) acc[j] = (v8f)(0.0f);

  const float* Arow = A + (size_t)rowc * K;
  for (int k = 0; k < K; k += 4) {
    v2f a = *(const v2f*)(Arow + k + khalf);
    const float* Bk0 = B + (size_t)(k + khalf) * N;
#pragma unroll
    for (int j = 0; j < 4; ++j) {
      int col = colb + j * 16;
      v2f b;
      b.x = Bk0[col];
      b.y = Bk0[N + col];
      acc[j] = __builtin_amdgcn_wmma_f32_16x16x4_f32(
          false, a, false, b, (short)0, acc[j], false, false);
    }
  }

#pragma unroll
  for (int j = 0; j < 4; ++j) {
    int col = colb + j * 16;
#pragma unroll
    for (int v = 0; v < 8; ++v) {
      int r = mt * 16 + v + ((lane >> 4) << 3);
      if (r < M) C[(size_t)r * N + col] = acc[j][v];
    }
  }
}

// ---------- attention coefficients, layer 1 (H=4, C=64) ----------
__global__ void gat_attn1(const float* __restrict__ xl,
                          const float* __restrict__ att_s,
                          const float* __restrict__ att_d,
                          float* __restrict__ as, float* __restrict__ ad, int N) {
  int tid = blockIdx.x * blockDim.x + threadIdx.x;
  if (tid >= N * 4) return;
  int n = tid >> 2, h = tid & 3;
  const float* xp = xl + (size_t)n * 256 + h * 64;
  const float* sp = att_s + h * 64;
  const float* dp = att_d + h * 64;
  float ss = 0.f, sd = 0.f;
#pragma unroll
  for (int c = 0; c < 64; c += 4) {
    float4 xv = *(const float4*)(xp + c);
    float4 sv = *(const float4*)(sp + c);
    float4 dv = *(const float4*)(dp + c);
    ss += xv.x * sv.x + xv.y * sv.y + xv.z * sv.z + xv.w * sv.w;
    sd += xv.x * dv.x + xv.y * dv.y + xv.z * dv.z + xv.w * dv.w;
  }
  as[tid] = ss;
  ad[tid] = sd;
}

// ---------- attention coefficients, layer 2 (H=1, C=128) ----------
__global__ void gat_attn2(const float* __restrict__ xl,
                          const float* __restrict__ att_s,
                          const float* __restrict__ att_d,
                          float* __restrict__ as, float* __restrict__ ad, int N) {
  int n = blockIdx.x * blockDim.x + threadIdx.x;
  if (n >= N) return;
  const float* xp = xl + (size_t)n * 128;
  float ss = 0.f, sd = 0.f;
#pragma unroll
  for (int c = 0; c < 128; c += 4) {
    float4 xv = *(const float4*)(xp + c);
    float4 sv = *(const float4*)(att_s + c);
    float4 dv = *(const float4*)(att_d + c);
    ss += xv.x * sv.x + xv.y * sv.y + xv.z * sv.z + xv.w * sv.w;
    sd += xv.x * dv.x + xv.y * dv.y + xv.z * dv.z + xv.w * dv.w;
  }
  as[n] = ss;
  ad[n] = sd;
}

// ---------- softmax pass 1: e = leaky_relu(a_src[s]+a_dst[d]); atomic max ----------
__global__ void gat_edge_pass1(const int* __restrict__ srcv, const int* __restrict__ dstv,
                               const float* __restrict__ as, const float* __restrict__ ad,
                               float* __restrict__ ebuf, unsigned* __restrict__ mkey,
                               int E, int ET, int H) {
  int tid = blockIdx.x * blockDim.x + threadIdx.x;
  if (tid >= ET * H) return;
  int eid = tid / H, h = tid - eid * H;
  int s, d;
  if (eid < E) { s = srcv[eid]; d = dstv[eid]; } else { s = d = eid - E; }
  float e = as[s * H + h] + ad[d * H + h];
  e = e > 0.f ? e : LRELU_SLOPE * e;
  ebuf[tid] = e;
  atomicMax(&mkey[d * H + h], fkey(e));
}

// ---------- softmax pass 2: e = exp(e - m[dst]); atomic add denom ----------
__global__ void gat_edge_pass2(const int* __restrict__ dstv,
                               float* __restrict__ ebuf,
                               const unsigned* __restrict__ mkey,
                               float* __restrict__ denom,
                               int E, int ET, int H) {
  int tid = blockIdx.x * blockDim.x + threadIdx.x;
  if (tid >= ET * H) return;
  int eid = tid / H, h = tid - eid * H;
  int d = (eid < E) ? dstv[eid] : (eid - E);
  float m = funkey(mkey[d * H + h]);
  float v = __expf(ebuf[tid] - m);
  ebuf[tid] = v;
  atomicAdd(&denom[d * H + h], v);
}

// ---------- aggregation layer 1: wave per edge, 256 elems (H=4,C=64) ----------
__global__ void gat_aggr1(const int* __restrict__ srcv, const int* __restrict__ dstv,
                          const float* __restrict__ xl, const float* __restrict__ ebuf,
                          const float* __restrict__ denom, float* __restrict__ outp,
                          int E, int ET) {
  int wid = (blockIdx.x * blockDim.x + threadIdx.x) >> 5;
  int lane = threadIdx.x & 31;
  if (wid >= ET) return;
  int s, d;
  if (wid < E) { s = srcv[wid]; d = dstv[wid]; } else { s = d = wid - E; }
  float alpha[4];
#pragma unroll
  for (int h = 0; h < 4; ++h)
    alpha[h] = ebuf[wid * 4 + h] / (denom[d * 4 + h] + 1e-16f);
  const float* xs = xl + (size_t)s * 256;
  float* od = outp + (size_t)d * 256;
#pragma unroll
  for (int j = 0; j < 8; ++j) {
    int idx = j * 32 + lane;            // == h*64 + c, contiguous & coalesced
    atomicAdd(&od[idx], alpha[j >> 1] * xs[idx]);
  }
}

// ---------- aggregation layer 2: wave per edge, 128 elems (H=1,C=128) ----------
__global__ void gat_aggr2(const int* __restrict__ srcv, const int* __restrict__ dstv,
                          const float* __restrict__ xl, const float* __restrict__ ebuf,
                          const float* __restrict__ denom, float* __restrict__ outp,
                          int E, int ET) {
  int wid = (blockIdx.x * blockDim.x + threadIdx.x) >> 5;
  int lane = threadIdx.x & 31;
  if (wid >= ET) return;
  int s, d;
  if (wid < E) { s = srcv[wid]; d = dstv[wid]; } else { s = d = wid - E; }
  float alpha = ebuf[wid] / (denom[d] + 1e-16f);
  const float* xs = xl + (size_t)s * 128;
  float* od = outp + (size_t)d * 128;
#pragma unroll
  for (int j = 0; j < 4; ++j) {
    int idx = j * 32 + lane;
    atomicAdd(&od[idx], alpha * xs[idx]);
  }
}

// ---------- bias + relu (layer 1, in place) ----------
__global__ void gat_bias_relu(float* __restrict__ h, const float* __restrict__ bias, int total) {
  int tid = blockIdx.x * blockDim.x + threadIdx.x;
  if (tid >= total) return;
  float v = h[tid] + bias[tid & 255];
  h[tid] = v > 0.f ? v : 0.f;
}

// ---------- global mean pool: accumulate ----------
__global__ void gat_pool_accum(const float* __restrict__ out2, const float* __restrict__ bias2,
                               const int* __restrict__ batch, float* __restrict__ pool,
                               float* __restrict__ cnt, int N) {
  int tid = blockIdx.x * blockDim.x + threadIdx.x;
  if (tid >= N * 128) return;
  int n = tid >> 7, c = tid & 127;
  int g = batch[n];
  atomicAdd(&pool[g * 128 + c], out2[tid] + bias2[c]);
  if (c == 0) atomicAdd(&cnt[g], 1.0f);
}

// ---------- global mean pool: finalize ----------
__global__ void gat_pool_final(const float* __restrict__ pool, const float* __restrict__ cnt,
                               float* __restrict__ outp, int total) {
  int tid = blockIdx.x * blockDim.x + threadIdx.x;
  if (tid >= total) return;
  outp[tid] = pool[tid] / fmaxf(cnt[tid >> 7], 1.0f);
}

static inline char* carve(char*& p, size_t bytes) {
  char* r = p;
  p += (bytes + 255) & ~(size_t)255;
  return r;
}

extern "C" void kernel_launch(void* const* d_in, const int* in_sizes, int n_in,
                              void* d_out, int out_size, void* d_ws, size_t ws_size,
                              hipStream_t stream) {
  const float* x        = (const float*)d_in[0];
  const int*   eidx     = (const int*)d_in[1];
  const int*   batch    = (const int*)d_in[2];
  const float* W1       = (const float*)d_in[3];
  const float* att_src1 = (const float*)d_in[4];
  const float* att_dst1 = (const float*)d_in[5];
  const float* bias1    = (const float*)d_in[6];
  const float* W2       = (const float*)d_in[7];
  const float* att_src2 = (const float*)d_in[8];
  const float* att_dst2 = (const float*)d_in[9];
  const float* bias2    = (const float*)d_in[10];
  float* out = (float*)d_out;

  const int N  = in_sizes[0] / 128;   // 50000
  const int E  = in_sizes[1] / 2;     // 800000
  const int ET = E + N;               // with self loops
  const int* srcv = eidx;
  const int* dstv = eidx + E;

  // ---- workspace layout ----
  char* p = (char*)d_ws;
  float* xl1 = (float*)carve(p, (size_t)N * 256 * 4);   // x @ W1
  float* xl2 = (float*)carve(p, (size_t)N * 128 * 4);   // h @ W2
  float* as1 = (float*)carve(p, (size_t)N * 4 * 4);
  float* ad1 = (float*)carve(p, (size_t)N * 4 * 4);
  float* as2 = (float*)carve(p, (size_t)N * 4);
  float* ad2 = (float*)carve(p, (size_t)N * 4);
  float* e1  = (float*)carve(p, (size_t)ET * 4 * 4);
  float* e2  = (float*)carve(p, (size_t)ET * 4);
  // zero-initialized region (contiguous)
  char* zstart = p;
  float*    out1   = (float*)carve(p, (size_t)N * 256 * 4);
  float*    out2   = (float*)carve(p, (size_t)N * 128 * 4);
  unsigned* mkey1  = (unsigned*)carve(p, (size_t)N * 4 * 4);
  float*    den1   = (float*)carve(p, (size_t)N * 4 * 4);
  unsigned* mkey2  = (unsigned*)carve(p, (size_t)N * 4);
  float*    den2   = (float*)carve(p, (size_t)N * 4);
  float*    pool   = (float*)carve(p, 8 * 128 * 4);
  float*    cnt    = (float*)carve(p, 8 * 4);
  size_t zbytes = (size_t)(p - zstart);
  hipMemsetAsync(zstart, 0, zbytes, stream);

  const int BLK = 256;
  int mtiles = (N + 15) >> 4;

  // ===== Layer 1 =====
  // GEMM1: xl1[N,256] = x[N,128] @ W1[128,256]
  {
    int waves = mtiles * (256 / 64);
    int blocks = (waves * 32 + BLK - 1) / BLK;
    gat_wmma_gemm_f32<<<blocks, BLK, 0, stream>>>(x, W1, xl1, N, 128, 256);
  }
  gat_attn1<<<(N * 4 + BLK - 1) / BLK, BLK, 0, stream>>>(xl1, att_src1, att_dst1, as1, ad1, N);
  gat_edge_pass1<<<(ET * 4 + BLK - 1) / BLK, BLK, 0, stream>>>(srcv, dstv, as1, ad1, e1, mkey1, E, ET, 4);
  gat_edge_pass2<<<(ET * 4 + BLK - 1) / BLK, BLK, 0, stream>>>(dstv, e1, mkey1, den1, E, ET, 4);
  gat_aggr1<<<(ET + 7) / 8, BLK, 0, stream>>>(srcv, dstv, xl1, e1, den1, out1, E, ET);
  gat_bias_relu<<<((size_t)N * 256 + BLK - 1) / BLK, BLK, 0, stream>>>(out1, bias1, N * 256);

  // ===== Layer 2 =====
  // GEMM2: xl2[N,128] = out1[N,256] @ W2[256,128]
  {
    int waves = mtiles * (128 / 64);
    int blocks = (waves * 32 + BLK - 1) / BLK;
    gat_wmma_gemm_f32<<<blocks, BLK, 0, stream>>>(out1, W2, xl2, N, 256, 128);
  }
  gat_attn2<<<(N + BLK - 1) / BLK, BLK, 0, stream>>>(xl2, att_src2, att_dst2, as2, ad2, N);
  gat_edge_pass1<<<(ET + BLK - 1) / BLK, BLK, 0, stream>>>(srcv, dstv, as2, ad2, e2, mkey2, E, ET, 1);
  gat_edge_pass2<<<(ET + BLK - 1) / BLK, BLK, 0, stream>>>(dstv, e2, mkey2, den2, E, ET, 1);
  gat_aggr2<<<(ET + 7) / 8, BLK, 0, stream>>>(srcv, dstv, xl2, e2, den2, out2, E, ET);

  // ===== Global mean pool =====
  gat_pool_accum<<<((size_t)N * 128 + BLK - 1) / BLK, BLK, 0, stream>>>(out2, bias2, batch, pool, cnt, N);
  gat_pool_final<<<(8 * 128 + BLK - 1) / BLK, BLK, 0, stream>>>(pool, cnt, out, 8 * 128);
}